// ZonalFunctions_23063974379633
// MI455X (gfx1250) — compile-verified
//
#include <hip/hip_runtime.h>
#include <cstdint>

// ---------------------------------------------------------------------------
// Zonal functions of the Lorentz group, l = 0..6, for B = 262144 points.
//
// Memory-bound streaming kernel (~300 MB traffic -> ~13us at 23.3 TB/s; the
// fully-unrolled CG recursion costs ~6us of VALU across the device, so HBM is
// the roof). Output (296 MB) exceeds the 192 MB L2 and is touched once, so all
// global traffic uses non-temporal hints.
//
// CDNA5 paths used:
//   * global_store_async_from_lds_b128 th:TH_STORE_NT (ASYNCcnt data mover,
//     fully-coalesced streaming drain of the LDS-staged output tiles)
//   * s_wait_dscnt / s_wait_asynccnt (split dependency counters)
//   * nontemporal load/store builtins -> th:TH_LOAD_NT / th:TH_STORE_NT
// WMMA is deliberately NOT used: the CG contraction is 2-banded (4 terms per
// output) with a per-point right operand, so dense 16x16 matrix tiles would
// waste >90% of the MACs on a kernel already at the HBM roofline.
// ---------------------------------------------------------------------------

#define MAX_ZF   6
#define ZF_PER   140                  // sum_{l=0..6} (l+1)^2
#define LANES    32                   // one wave per block (wave32)
#define INV_SQRT2 0.70710678118654752440f

typedef __attribute__((ext_vector_type(4))) float v4f;  // native vector type
                                                        // (accepted by the
                                                        // nontemporal builtins)

// Compute Z_L from Z_{L-1} (size LxL) and P (2x2), write Z_L into the
// per-thread LDS slice, then recurse. Fully unrolled: every array index is a
// constant after unrolling, so zr/zi/nr/ni live in VGPRs and every CG
// coefficient folds to a literal.
template <int L>
__device__ __forceinline__ void zonal_step(const float (&zr)[L][L],
                                           const float (&zi)[L][L],
                                           const float (&pr)[2][2],
                                           const float (&pi)[2][2],
                                           float* __restrict__ mRe,
                                           float* __restrict__ mIm) {
  float nr[L + 1][L + 1];
  float ni[L + 1][L + 1];
  const float sc = sqrtf(2.0f * (float)L / (float)(L + 1));  // folded to literal

#pragma unroll
  for (int M = 0; M <= L; ++M) {
    const float aM = sqrtf((float)M / (float)L);        // C[M, M-1, 1]
    const float bM = sqrtf((float)(L - M) / (float)L);  // C[M, M,   0]
#pragma unroll
    for (int N = 0; N <= L; ++N) {
      const float aN = sqrtf((float)N / (float)L);
      const float bN = sqrtf((float)(L - N) / (float)L);
      float ar = 0.0f, ai = 0.0f;
      if (M >= 1 && N >= 1) {          // Z[M-1][N-1] * P[1][1]
        const float c = sc * aM * aN;  // compile-time literal
        ar += c * (zr[M-1][N-1] * pr[1][1] - zi[M-1][N-1] * pi[1][1]);
        ai += c * (zr[M-1][N-1] * pi[1][1] + zi[M-1][N-1] * pr[1][1]);
      }
      if (M >= 1 && N <= L - 1) {      // Z[M-1][N] * P[1][0]
        const float c = sc * aM * bN;
        ar += c * (zr[M-1][N] * pr[1][0] - zi[M-1][N] * pi[1][0]);
        ai += c * (zr[M-1][N] * pi[1][0] + zi[M-1][N] * pr[1][0]);
      }
      if (M <= L - 1 && N >= 1) {      // Z[M][N-1] * P[0][1]
        const float c = sc * bM * aN;
        ar += c * (zr[M][N-1] * pr[0][1] - zi[M][N-1] * pi[0][1]);
        ai += c * (zr[M][N-1] * pi[0][1] + zi[M][N-1] * pr[0][1]);
      }
      if (M <= L - 1 && N <= L - 1) {  // Z[M][N] * P[0][0]
        const float c = sc * bM * bN;
        ar += c * (zr[M][N] * pr[0][0] - zi[M][N] * pi[0][0]);
        ai += c * (zr[M][N] * pi[0][0] + zi[M][N] * pr[0][0]);
      }
      nr[M][N] = ar;
      ni[M][N] = ai;
    }
  }

  constexpr int off = L * (L + 1) * (2 * L + 1) / 6;  // sum_{k=1..L} k^2
#pragma unroll
  for (int M = 0; M <= L; ++M)
#pragma unroll
    for (int N = 0; N <= L; ++N) {
      mRe[off + M * (L + 1) + N] = nr[M][N];
      mIm[off + M * (L + 1) + N] = ni[M][N];
    }

  if constexpr (L < MAX_ZF) zonal_step<L + 1>(nr, ni, pr, pi, mRe, mIm);
}

__global__ __launch_bounds__(LANES) void zonal_functions_kernel(
    const float* __restrict__ pos, float* __restrict__ out, int B) {
  // Per-wave staging tiles: lane-major in LDS, drained with coalesced async
  // stores (global layout == LDS layout, so the drain is a flat memcpy).
  __shared__ __align__(16) float sRe[LANES * ZF_PER];  // 17.5 KB
  __shared__ __align__(16) float sIm[LANES * ZF_PER];  // 17.5 KB

  const int lane = threadIdx.x;          // wave32: lane == lane id
  const int b = blockIdx.x * LANES + lane;

  // Streaming read-once input -> non-temporal load hint.
  const v4f p = __builtin_nontemporal_load((const v4f*)(pos + (size_t)b * 4));
  const float t = p.x, x = p.y, y = p.z, z = p.w;

  // (1,1) irrep as 2x2 complex matrix (row-major reshape of
  //  [t, (x-iy)/sqrt2, z, (-x-iy)/sqrt2])
  const float pr[2][2] = {{t, x * INV_SQRT2}, {z, -x * INV_SQRT2}};
  const float pi[2][2] = {{0.0f, -y * INV_SQRT2}, {0.0f, -y * INV_SQRT2}};

  float* mRe = sRe + lane * ZF_PER;      // 560 B, 16B-aligned
  float* mIm = sIm + lane * ZF_PER;

  // l = 0: ones in BOTH real and imaginary parts (matches reference).
  mRe[0] = 1.0f;      mIm[0] = 1.0f;
  // l = 1: p_rep itself.
  mRe[1] = pr[0][0];  mRe[2] = pr[0][1];  mRe[3] = pr[1][0];  mRe[4] = pr[1][1];
  mIm[1] = pi[0][0];  mIm[2] = pi[0][1];  mIm[3] = pi[1][0];  mIm[4] = pi[1][1];

  // l = 2..6 recursion (Z_1 == P).
  zonal_step<2>(pr, pi, pr, pi, mRe, mIm);

  // Minkowski norm outputs (lane-consecutive -> naturally coalesced).
  const float ns = t * t - x * x - y * y - z * z;
  const float nrm = (ns != 0.0f) ? ns / sqrtf(fabsf(ns)) : 0.0f;
  float* tail = out + 2ull * (size_t)B * ZF_PER;
  __builtin_nontemporal_store(nrm, tail + b);
  __builtin_nontemporal_store(ns, tail + (size_t)B + b);

  // --- CDNA5 async drain: LDS -> global, fully coalesced ------------------
  // Make sure this wave's ds_store writes have landed before the async data
  // mover reads LDS (async LDS reads are not ordered with DScnt by hardware).
  asm volatile("s_wait_dscnt 0" ::: "memory");

  float* gRe = out + (size_t)blockIdx.x * (LANES * ZF_PER);
  float* gIm = out + (size_t)B * ZF_PER + (size_t)blockIdx.x * (LANES * ZF_PER);

  // 4480 floats per part = 35 iterations of 32 lanes x float4.
  // NT store hint: written-once stream, keep it out of L2 (output > L2 size).
  for (int it = 0; it < (LANES * ZF_PER) / (LANES * 4); ++it) {
    const int idx = it * (LANES * 4) + lane * 4;
    const float* dstRe = gRe + idx;
    const float* dstIm = gIm + idx;
    // Low 32 bits of a shared-memory flat pointer == LDS byte offset.
    const uint32_t srcRe = (uint32_t)(uintptr_t)(sRe + idx);
    const uint32_t srcIm = (uint32_t)(uintptr_t)(sIm + idx);
    asm volatile("global_store_async_from_lds_b128 %0, %1, off th:TH_STORE_NT"
                 :: "v"(dstRe), "v"(srcRe) : "memory");
    asm volatile("global_store_async_from_lds_b128 %0, %1, off th:TH_STORE_NT"
                 :: "v"(dstIm), "v"(srcIm) : "memory");
  }
  // Drain ASYNCcnt before wave exit (s_endpgm would also imply s_wait_idle).
  asm volatile("s_wait_asynccnt 0" ::: "memory");
}

extern "C" void kernel_launch(void* const* d_in, const int* in_sizes, int n_in,
                              void* d_out, int out_size, void* d_ws, size_t ws_size,
                              hipStream_t stream) {
  (void)n_in; (void)out_size; (void)d_ws; (void)ws_size;
  const float* pos = (const float*)d_in[0];
  float* out = (float*)d_out;
  const int B = in_sizes[0] / 4;        // 262144
  const int blocks = B / LANES;         // 8192 single-wave workgroups
  zonal_functions_kernel<<<dim3(blocks), dim3(LANES), 0, stream>>>(pos, out, B);
}